// GlobalAttentionPooling_15066745274947
// MI455X (gfx1250) — compile-verified
//
#include <hip/hip_runtime.h>
#include <math.h>

// GraphConv + global attention pooling for MI455X (gfx1250, wave32).
// Scatters are L2-resident atomic-add bound (everything fits in 192MB L2);
// GEMM1 uses fp32 WMMA (V_WMMA_F32_16X16X4_F32) with K-pair-interleaved
// weights in LDS so every B operand is one aligned ds_load_b64.

typedef __attribute__((ext_vector_type(2))) float v2f;
typedef __attribute__((ext_vector_type(8))) float v8f;

#define DFEAT 128
#define L1DIM 64

__device__ __forceinline__ void atomic_add_f32(float* p, float v) {
  __hip_atomic_fetch_add(p, v, __ATOMIC_RELAXED, __HIP_MEMORY_SCOPE_AGENT);
}

// leaky_relu(v, 0.2) == max(v, 0.2*v) since slope < 1  -> v_mul + v_max_f32
__device__ __forceinline__ float leaky02(float v) { return fmaxf(v, 0.2f * v); }

// ---------------- zero fill ----------------
__global__ void zero_f32(float* __restrict__ p, size_t n) {
  size_t i = (size_t)blockIdx.x * blockDim.x + threadIdx.x;
  size_t stride = (size_t)gridDim.x * blockDim.x;
  for (; i < n; i += stride) p[i] = 0.0f;
}

// ---------------- scatter-add, 128-wide rows: one wave per edge ----------------
__global__ void scatter_add_d128(const float* __restrict__ x,
                                 const int* __restrict__ src,
                                 const int* __restrict__ dst,
                                 float* __restrict__ agg, int nedge) {
  int wave = (int)(((size_t)blockIdx.x * blockDim.x + threadIdx.x) >> 5);
  int lane = threadIdx.x & 31;
  if (wave >= nedge) return;
  int s = src[wave];
  int d = dst[wave];
  const float4* xs = (const float4*)(x + (size_t)s * DFEAT);
  float4 v = xs[lane];                      // lane covers 4 consecutive floats
  float* ap = agg + (size_t)d * DFEAT + lane * 4;
  atomic_add_f32(ap + 0, v.x);
  atomic_add_f32(ap + 1, v.y);
  atomic_add_f32(ap + 2, v.z);
  atomic_add_f32(ap + 3, v.w);
}

// ---------------- scatter-add, 64-wide rows: one wave per edge ----------------
__global__ void scatter_add_d64(const float* __restrict__ h,
                                const int* __restrict__ src,
                                const int* __restrict__ dst,
                                float* __restrict__ agg, int nedge) {
  int wave = (int)(((size_t)blockIdx.x * blockDim.x + threadIdx.x) >> 5);
  int lane = threadIdx.x & 31;
  if (wave >= nedge) return;
  int s = src[wave];
  int d = dst[wave];
  const float2* hs = (const float2*)(h + (size_t)s * L1DIM);
  float2 v = hs[lane];
  float* ap = agg + (size_t)d * L1DIM + lane * 2;
  atomic_add_f32(ap + 0, v.x);
  atomic_add_f32(ap + 1, v.y);
}

// ---------------- fused dual-GEMM + bias + leaky_relu via fp32 WMMA ----------------
// h[n, l] = leaky_relu( sum_k agg1[n,k]*W1_rel[l,k] + x[n,k]*W1_root[l,k] + b1[l] )
// One wave computes a 16-node x 64-output tile.
// LDS layout is K-pair interleaved: sW[kp][l] = {W[l][2kp], W[l][2kp+1]} so the
// two K-values a lane feeds into the A/B VGPR pair are one aligned b64 load.
__global__ __launch_bounds__(256)
void gemm1_wmma(const float* __restrict__ agg1, const float* __restrict__ x,
                const float* __restrict__ W1_rel, const float* __restrict__ W1_root,
                const float* __restrict__ b1, float* __restrict__ h, int n_nodes) {
  __shared__ float2 sWrel[DFEAT / 2][L1DIM];    // 32 KB
  __shared__ float2 sWroot[DFEAT / 2][L1DIM];   // 32 KB

  // Stage weights: i enumerates (l, kp); consecutive threads -> consecutive kp
  // -> coalesced 8B global reads along each weight row.
  for (int i = threadIdx.x; i < L1DIM * (DFEAT / 2); i += blockDim.x) {
    int l = i >> 6;         // 0..63  output index
    int kp = i & 63;        // 0..63  K pair index
    sWrel[kp][l]  = *(const float2*)&W1_rel[(size_t)l * DFEAT + 2 * kp];
    sWroot[kp][l] = *(const float2*)&W1_root[(size_t)l * DFEAT + 2 * kp];
  }
  __syncthreads();

  int waveInBlock = threadIdx.x >> 5;
  int lane = threadIdx.x & 31;
  int tile = blockIdx.x * (blockDim.x >> 5) + waveInBlock;
  int numTiles = (n_nodes + 15) >> 4;
  if (tile >= numTiles) return;          // wave-uniform: EXEC stays all-ones for WMMA
  int m0 = tile << 4;

  int half = lane >> 4;      // 0: K pair {0,1}   1: K pair {2,3}
  int l16  = lane & 15;
  int kb   = half * 2;       // element offset of this half's K pair

  int arow = m0 + l16;
  if (arow >= n_nodes) arow = n_nodes - 1;   // clamp; store is guarded below
  const float* aggRow = agg1 + (size_t)arow * DFEAT;
  const float* xRow   = x    + (size_t)arow * DFEAT;

  // C/D layout: VGPR j -> row m0 + j + 8*half, col = nt*16 + l16.
  // Bias depends only on the column -> same value in all 8 accumulator VGPRs.
  v8f acc[4];
  for (int nt = 0; nt < 4; ++nt) {
    float bv = b1[nt * 16 + l16];
    for (int j = 0; j < 8; ++j) acc[nt][j] = bv;
  }

  for (int k = 0; k < DFEAT; k += 4) {
    int kp = (k >> 1) + half;            // this half's K-pair row
    // A tile (16x4 fp32): VGPR0 = K=kb, VGPR1 = K=kb+1 -- one b64 load each
    float2 av = *(const float2*)(aggRow + k + kb);
    float2 xv = *(const float2*)(xRow + k + kb);
    v2f a_agg; a_agg.x = av.x; a_agg.y = av.y;
    v2f a_x;   a_x.x   = xv.x; a_x.y   = xv.y;
#pragma unroll
    for (int nt = 0; nt < 4; ++nt) {
      int l = nt * 16 + l16;
      float2 br = sWrel[kp][l];          // single ds_load_b64
      float2 bo = sWroot[kp][l];         // single ds_load_b64
      v2f b_rel;  b_rel.x = br.x;  b_rel.y = br.y;
      v2f b_root; b_root.x = bo.x; b_root.y = bo.y;
      acc[nt] = __builtin_amdgcn_wmma_f32_16x16x4_f32(
          false, a_agg, false, b_rel, (short)0, acc[nt], false, false);
      acc[nt] = __builtin_amdgcn_wmma_f32_16x16x4_f32(
          false, a_x, false, b_root, (short)0, acc[nt], false, false);
    }
  }

  // leaky_relu(0.2) + store.  Fast path for full tiles: no per-element exec
  // masking (uniform branch; all tiles are full when 16 | n_nodes).
  if (m0 + 16 <= n_nodes) {
#pragma unroll
    for (int nt = 0; nt < 4; ++nt) {
      int col = nt * 16 + l16;
#pragma unroll
      for (int j = 0; j < 8; ++j) {
        int row = m0 + j + half * 8;
        h[(size_t)row * L1DIM + col] = leaky02(acc[nt][j]);
      }
    }
  } else {
#pragma unroll
    for (int nt = 0; nt < 4; ++nt) {
      int col = nt * 16 + l16;
#pragma unroll
      for (int j = 0; j < 8; ++j) {
        int row = m0 + j + half * 8;
        if (row < n_nodes) h[(size_t)row * L1DIM + col] = leaky02(acc[nt][j]);
      }
    }
  }
}

// ---------------- gate[n] = agg2[n,:].W2_rel + b2 + h[n,:].W2_root ----------------
// Wave per node: lanes cover the 64-wide rows with float2 loads (coalesced),
// then a 5-step shfl_xor wave32 reduction.
__global__ void gate_kernel(const float* __restrict__ agg2, const float* __restrict__ h,
                            const float* __restrict__ W2_rel, const float* __restrict__ b2,
                            const float* __restrict__ W2_root,
                            float* __restrict__ gate, int n) {
  int wave = (int)(((size_t)blockIdx.x * blockDim.x + threadIdx.x) >> 5);
  int lane = threadIdx.x & 31;
  if (wave >= n) return;
  const float2* a  = (const float2*)(agg2 + (size_t)wave * L1DIM);
  const float2* hr = (const float2*)(h + (size_t)wave * L1DIM);
  const float2* wr = (const float2*)W2_rel;
  const float2* wo = (const float2*)W2_root;
  float2 av = a[lane], hv = hr[lane], wv = wr[lane], ov = wo[lane];
  float s = av.x * wv.x + av.y * wv.y + hv.x * ov.x + hv.y * ov.y;
#pragma unroll
  for (int m = 16; m >= 1; m >>= 1) s += __shfl_xor(s, m, 32);
  if (lane == 0) gate[wave] = s + b2[0];
}

// ---------------- softmax pass 1: block maxima ----------------
__global__ void reduce_max(const float* __restrict__ g, float* __restrict__ pmax, int n) {
  __shared__ float sm[256];
  float m = -INFINITY;
  for (int i = blockIdx.x * blockDim.x + threadIdx.x; i < n;
       i += gridDim.x * blockDim.x)
    m = fmaxf(m, g[i]);
  sm[threadIdx.x] = m;
  __syncthreads();
  for (int s = 128; s > 0; s >>= 1) {
    if ((int)threadIdx.x < s) sm[threadIdx.x] = fmaxf(sm[threadIdx.x], sm[threadIdx.x + s]);
    __syncthreads();
  }
  if (threadIdx.x == 0) pmax[blockIdx.x] = sm[0];
}

__global__ void final_max(const float* __restrict__ pmax, float* __restrict__ scal_max) {
  __shared__ float sm[256];
  sm[threadIdx.x] = pmax[threadIdx.x];
  __syncthreads();
  for (int s = 128; s > 0; s >>= 1) {
    if ((int)threadIdx.x < s) sm[threadIdx.x] = fmaxf(sm[threadIdx.x], sm[threadIdx.x + s]);
    __syncthreads();
  }
  if (threadIdx.x == 0) scal_max[0] = sm[0];
}

// ---------------- softmax pass 2: e[i] = exp(g[i]-max); block sums ----------------
__global__ void exp_partial(const float* __restrict__ g, const float* __restrict__ scal_max,
                            float* __restrict__ ex, float* __restrict__ psum, int n) {
  __shared__ float sm[256];
  float mx = scal_max[0];
  float s = 0.0f;
  for (int i = blockIdx.x * blockDim.x + threadIdx.x; i < n;
       i += gridDim.x * blockDim.x) {
    float v = __expf(g[i] - mx);
    ex[i] = v;
    s += v;
  }
  sm[threadIdx.x] = s;
  __syncthreads();
  for (int st = 128; st > 0; st >>= 1) {
    if ((int)threadIdx.x < st) sm[threadIdx.x] += sm[threadIdx.x + st];
    __syncthreads();
  }
  if (threadIdx.x == 0) psum[blockIdx.x] = sm[0];
}

// single block: finalize sum AND zero d_out (harness poisons it)
__global__ void final_sum(const float* __restrict__ psum, float* __restrict__ scal_sum,
                          float* __restrict__ out) {
  __shared__ float sm[256];
  sm[threadIdx.x] = psum[threadIdx.x];
  __syncthreads();
  for (int s = 128; s > 0; s >>= 1) {
    if ((int)threadIdx.x < s) sm[threadIdx.x] += sm[threadIdx.x + s];
    __syncthreads();
  }
  if (threadIdx.x == 0) scal_sum[0] = sm[0];
  if (threadIdx.x < DFEAT) out[threadIdx.x] = 0.0f;
}

// ---------------- out[d] = sum_n (e[n]/sum) * x[n,d] ----------------
#define NODES_PER_BLOCK 512
__global__ void weighted_colsum(const float* __restrict__ x, const float* __restrict__ ex,
                                const float* __restrict__ scal_sum,
                                float* __restrict__ out, int n) {
  float inv = 1.0f / scal_sum[0];
  int d = threadIdx.x & 127;   // column
  int p = threadIdx.x >> 7;    // 0/1: node parity within block chunk
  int start = blockIdx.x * NODES_PER_BLOCK;
  int end = start + NODES_PER_BLOCK;
  if (end > n) end = n;
  float acc = 0.0f;
  for (int i = start + p; i < end; i += 2)
    acc += ex[i] * x[(size_t)i * DFEAT + d];
  atomic_add_f32(&out[d], acc * inv);
}

extern "C" void kernel_launch(void* const* d_in, const int* in_sizes, int n_in,
                              void* d_out, int out_size, void* d_ws, size_t ws_size,
                              hipStream_t stream) {
  const float* x       = (const float*)d_in[0];
  const int*   ei      = (const int*)d_in[1];
  const float* W1_rel  = (const float*)d_in[2];
  const float* b1      = (const float*)d_in[3];
  const float* W1_root = (const float*)d_in[4];
  const float* W2_rel  = (const float*)d_in[5];
  const float* b2      = (const float*)d_in[6];
  const float* W2_root = (const float*)d_in[7];
  float* out = (float*)d_out;

  const int n = in_sizes[0] / DFEAT;   // nodes
  const int e = in_sizes[1] / 2;       // edges
  const int* src = ei;
  const int* dst = ei + e;

  // Workspace layout (floats). agg1 [n*128] is dead after gemm1, so the
  // later buffers alias its region:
  //   phase A/B: agg1 = ws[0 .. n*128) ; h = ws[n*128 .. n*192)
  //   phase C+:  agg2 = ws[0 .. n*64)  ; gate, ex, partials, scalars follow
  float* ws   = (float*)d_ws;
  float* agg1 = ws;
  float* h    = ws + (size_t)n * DFEAT;
  float* agg2 = ws;
  float* gate = ws + (size_t)n * L1DIM;
  float* ex   = gate + n;
  float* pmax = ex + n;       // 256
  float* psum = pmax + 256;   // 256
  float* scal = psum + 256;   // [0]=max, [1]=sum

  const int sblocks = (int)(((size_t)e * 32 + 255) / 256);  // one wave per edge
  const int tiles = (n + 15) >> 4;

  // 1) agg1 = segment_sum(x[src] -> dst)
  zero_f32<<<2048, 256, 0, stream>>>(agg1, (size_t)n * DFEAT);
  scatter_add_d128<<<sblocks, 256, 0, stream>>>(x, src, dst, agg1, e);

  // 2) h = leaky_relu(agg1 @ W1_rel.T + b1 + x @ W1_root.T)   [WMMA fp32]
  gemm1_wmma<<<(tiles + 7) / 8, 256, 0, stream>>>(agg1, x, W1_rel, W1_root, b1, h, n);

  // 3) agg2 = segment_sum(h[src] -> dst)   (agg1 region reused)
  zero_f32<<<2048, 256, 0, stream>>>(agg2, (size_t)n * L1DIM);
  scatter_add_d64<<<sblocks, 256, 0, stream>>>(h, src, dst, agg2, e);

  // 4) gate = agg2 @ W2_rel.T + b2 + h @ W2_root.T   (wave per node)
  gate_kernel<<<(int)(((size_t)n * 32 + 255) / 256), 256, 0, stream>>>(
      agg2, h, W2_rel, b2, W2_root, gate, n);

  // 5) softmax over nodes
  reduce_max<<<256, 256, 0, stream>>>(gate, pmax, n);
  final_max<<<1, 256, 0, stream>>>(pmax, scal);
  exp_partial<<<256, 256, 0, stream>>>(gate, scal, ex, psum, n);
  final_sum<<<1, 256, 0, stream>>>(psum, scal + 1, out);   // also zeroes out[]

  // 6) out = softmax(gate).T @ x
  weighted_colsum<<<(n + NODES_PER_BLOCK - 1) / NODES_PER_BLOCK, 256, 0, stream>>>(
      x, ex, scal + 1, out, n);
}